// GraphConv_67353677136469
// MI455X (gfx1250) — compile-verified
//
#include <hip/hip_runtime.h>

typedef __attribute__((ext_vector_type(2))) float v2f;
typedef __attribute__((ext_vector_type(8))) float v8f;

#define SEED1 0x9E3779B9u
#define SEED2 0x7F4A7C15u

// -------- deterministic dropout mask: returns 2.0 (keep, 1/(1-p)) or 0.0 -----
__device__ __forceinline__ float drop_mask(unsigned idx, unsigned seed) {
    unsigned h = idx * 2654435761u ^ seed;
    h ^= h >> 16; h *= 0x85ebca6bu;
    h ^= h >> 13; h *= 0xc2b2ae35u;
    h ^= h >> 16;
    return (h & 1u) ? 2.0f : 0.0f;
}

__device__ __forceinline__ v8f wmma_f32(v2f a, v2f b, v8f c) {
    // D = A(16x4,f32) * B(4x16,f32) + C(16x16,f32)
    return __builtin_amdgcn_wmma_f32_16x16x4_f32(
        false, a, false, b, (short)0, c, false, false);
}

// ---------------------------------------------------------------------------
__global__ __launch_bounds__(256) void zero_k(float* p, long n) {
    long i = (long)blockIdx.x * blockDim.x + threadIdx.x;
    long stride = (long)gridDim.x * blockDim.x;
    for (; i < n; i += stride) p[i] = 0.0f;
}

__global__ __launch_bounds__(256) void count_k(const int* __restrict__ dst,
                                               float* __restrict__ cnt, int e) {
    int i = blockIdx.x * blockDim.x + threadIdx.x;
    if (i < e) atomicAdd(&cnt[dst[i]], 1.0f);
}

__global__ __launch_bounds__(256) void rinv_k(const float* __restrict__ cnt,
                                              float* __restrict__ rinv, int n) {
    int i = blockIdx.x * blockDim.x + threadIdx.x;
    if (i < n) rinv[i] = 1.0f / fmaxf(cnt[i], 1.0f);
}

// scatter-add feat[src] into agg[dst]; d = feature width (multiple of 4)
__global__ __launch_bounds__(256) void scatter_k(const float* __restrict__ feat,
                                                 const int* __restrict__ src,
                                                 const int* __restrict__ dst,
                                                 float* __restrict__ agg,
                                                 int e, int d) {
    int nchunk = d >> 2;
    long t = (long)blockIdx.x * blockDim.x + threadIdx.x;
    if (t >= (long)e * nchunk) return;
    int edge = (int)(t / nchunk);
    int c    = (int)(t % nchunk) << 2;
    const float4 v = *(const float4*)(feat + (long)src[edge] * d + c);
    float* a = agg + (long)dst[edge] * d + c;
    atomicAdd(a + 0, v.x);
    atomicAdd(a + 1, v.y);
    atomicAdd(a + 2, v.z);
    atomicAdd(a + 3, v.w);
}

// -------- layer 1: h = drop(lrelu(mean@Wl + b + x@Wr))  [64 -> 32] ----------
// LDS weights stored K-pair interleaved: sW[k/2][col] = {W[k][col], W[k+1][col]}
// so each B fragment is ONE aligned ds_load_b64.
__global__ __launch_bounds__(256) void sage1_k(const float* __restrict__ x,
                                               const float* __restrict__ agg,
                                               const float* __restrict__ rinv,
                                               const float* __restrict__ Wl,
                                               const float* __restrict__ Wr,
                                               const float* __restrict__ bias,
                                               float* __restrict__ h, int n) {
    __shared__ v2f sWl[32 * 32];   // [kpair=32][col=32]
    __shared__ v2f sWr[32 * 32];
    {
        float* pl = (float*)sWl;
        float* pr = (float*)sWr;
        for (int i = threadIdx.x; i < 64 * 32; i += blockDim.x) {
            const int k = i >> 5, c = i & 31;
            const int di = ((k >> 1) * 32 + c) * 2 + (k & 1);
            pl[di] = Wl[i];
            pr[di] = Wr[i];
        }
    }
    __syncthreads();

    const int lane = threadIdx.x & 31;
    const int wave = threadIdx.x >> 5;
    const int tile = blockIdx.x * 8 + wave;
    const int row0 = tile * 16;
    if (row0 >= n) return;

    const int lm   = lane & 15;        // A: row within tile / B: column
    const int kh   = lane >> 4;        // K-pair sub-offset (0 or 1)
    const int row  = row0 + lm;
    const float rs = rinv[row];
    const float* xr = x   + (long)row * 64;
    const float* ar = agg + (long)row * 64;

    v8f acc0 = {}; // output cols 0..15
    v8f acc1 = {}; // output cols 16..31
    #pragma unroll
    for (int k0 = 0; k0 < 64; k0 += 4) {
        const int k  = k0 + kh * 2;          // element K offset for A
        const int kp = (k0 >> 1) + kh;       // K-pair index for B
        v2f aM, aX;
        aM.x = ar[k] * rs;  aM.y = ar[k + 1] * rs;
        aX.x = xr[k];       aX.y = xr[k + 1];
        const v2f bL0 = sWl[kp * 32 + lm];
        const v2f bL1 = sWl[kp * 32 + 16 + lm];
        const v2f bR0 = sWr[kp * 32 + lm];
        const v2f bR1 = sWr[kp * 32 + 16 + lm];
        acc0 = wmma_f32(aM, bL0, acc0);
        acc0 = wmma_f32(aX, bR0, acc0);
        acc1 = wmma_f32(aM, bL1, acc1);
        acc1 = wmma_f32(aX, bR1, acc1);
    }

    const float bias0 = bias[lm];
    const float bias1 = bias[16 + lm];
    #pragma unroll
    for (int v = 0; v < 8; ++v) {
        const int r = row0 + v + kh * 8;
        float v0 = acc0[v] + bias0;
        v0 = (v0 > 0.0f) ? v0 : 0.01f * v0;
        v0 *= drop_mask((unsigned)(r * 32 + lm), SEED1);
        h[(long)r * 32 + lm] = v0;
        float v1 = acc1[v] + bias1;
        v1 = (v1 > 0.0f) ? v1 : 0.01f * v1;
        v1 *= drop_mask((unsigned)(r * 32 + 16 + lm), SEED1);
        h[(long)r * 32 + 16 + lm] = v1;
    }
}

// -------- layer 2: out = l2norm(drop(mean2@Wl + b + h@Wr))  [32 -> 64] ------
__global__ __launch_bounds__(256) void sage2_k(const float* __restrict__ hfeat,
                                               const float* __restrict__ agg,
                                               const float* __restrict__ rinv,
                                               const float* __restrict__ Wl,
                                               const float* __restrict__ Wr,
                                               const float* __restrict__ bias,
                                               float* __restrict__ out, int n) {
    __shared__ v2f sWl[16 * 64];   // [kpair=16][col=64]
    __shared__ v2f sWr[16 * 64];
    {
        float* pl = (float*)sWl;
        float* pr = (float*)sWr;
        for (int i = threadIdx.x; i < 32 * 64; i += blockDim.x) {
            const int k = i >> 6, c = i & 63;
            const int di = ((k >> 1) * 64 + c) * 2 + (k & 1);
            pl[di] = Wl[i];
            pr[di] = Wr[i];
        }
    }
    __syncthreads();

    const int lane = threadIdx.x & 31;
    const int wave = threadIdx.x >> 5;
    const int tile = blockIdx.x * 8 + wave;
    const int row0 = tile * 16;
    if (row0 >= n) return;

    const int lm   = lane & 15;
    const int kh   = lane >> 4;
    const int row  = row0 + lm;
    const float rs = rinv[row];
    const float* hr = hfeat + (long)row * 32;
    const float* ar = agg   + (long)row * 32;

    v8f acc[4] = {v8f{}, v8f{}, v8f{}, v8f{}};
    #pragma unroll
    for (int k0 = 0; k0 < 32; k0 += 4) {
        const int k  = k0 + kh * 2;
        const int kp = (k0 >> 1) + kh;
        v2f aM, aH;
        aM.x = ar[k] * rs;  aM.y = ar[k + 1] * rs;
        aH.x = hr[k];       aH.y = hr[k + 1];
        #pragma unroll
        for (int t = 0; t < 4; ++t) {
            const v2f bL = sWl[kp * 64 + t * 16 + lm];
            const v2f bR = sWr[kp * 64 + t * 16 + lm];
            acc[t] = wmma_f32(aM, bL, acc[t]);
            acc[t] = wmma_f32(aH, bR, acc[t]);
        }
    }

    float sq[8] = {0, 0, 0, 0, 0, 0, 0, 0};
    #pragma unroll
    for (int t = 0; t < 4; ++t) {
        const float bc = bias[t * 16 + lm];
        #pragma unroll
        for (int v = 0; v < 8; ++v) {
            const int r = row0 + v + kh * 8;
            float val = acc[t][v] + bc;
            val *= drop_mask((unsigned)(r * 64 + t * 16 + lm), SEED2);
            acc[t][v] = val;
            sq[v] += val * val;
        }
    }
    // reduce sum-of-squares across the 16 lanes that share each row
    #pragma unroll
    for (int v = 0; v < 8; ++v) {
        float s = sq[v];
        s += __shfl_xor(s, 1, 16);
        s += __shfl_xor(s, 2, 16);
        s += __shfl_xor(s, 4, 16);
        s += __shfl_xor(s, 8, 16);
        sq[v] = 1.0f / fmaxf(sqrtf(s), 1e-12f);
    }
    #pragma unroll
    for (int t = 0; t < 4; ++t) {
        #pragma unroll
        for (int v = 0; v < 8; ++v) {
            const int r = row0 + v + kh * 8;
            out[(long)r * 64 + t * 16 + lm] = acc[t][v] * sq[v];
        }
    }
}

// ---------------------------------------------------------------------------
extern "C" void kernel_launch(void* const* d_in, const int* in_sizes, int n_in,
                              void* d_out, int out_size, void* d_ws, size_t ws_size,
                              hipStream_t stream) {
    const float* x   = (const float*)d_in[0];
    const int*   ei  = (const int*)d_in[1];
    const float* W1l = (const float*)d_in[2];
    const float* b1  = (const float*)d_in[3];
    const float* W1r = (const float*)d_in[4];
    const float* W2l = (const float*)d_in[5];
    const float* b2  = (const float*)d_in[6];
    const float* W2r = (const float*)d_in[7];
    float* out = (float*)d_out;

    const int n = in_sizes[0] / 64;   // 100000
    const int e = in_sizes[1] / 2;    // 1600000
    const int* src = ei;
    const int* dst = ei + e;

    // workspace layout (floats)
    float* ws   = (float*)d_ws;
    float* cnt  = ws;                       // n
    float* rinv = ws + (long)n;             // n
    float* agg1 = ws + (long)2 * n;         // 64n
    float* agg2 = ws + (long)66 * n;        // 32n
    float* hbuf = ws + (long)98 * n;        // 32n

    // 1) zero cnt + rinv + agg1 + agg2 (first 98n floats)
    {
        long nz = (long)98 * n;
        zero_k<<<2048, 256, 0, stream>>>(ws, nz);
    }
    // 2) degree counts
    count_k<<<(e + 255) / 256, 256, 0, stream>>>(dst, cnt, e);
    // 3) scatter x into agg1 (d=64)
    {
        long threads = (long)e * 16;
        scatter_k<<<(int)((threads + 255) / 256), 256, 0, stream>>>(x, src, dst, agg1, e, 64);
    }
    // 4) reciprocal degrees
    rinv_k<<<(n + 255) / 256, 256, 0, stream>>>(cnt, rinv, n);
    // 5) layer 1 (WMMA): h = drop(lrelu(mean@W1l + b1 + x@W1r))
    {
        int tiles = (n + 15) / 16;
        sage1_k<<<(tiles + 7) / 8, 256, 0, stream>>>(x, agg1, rinv, W1l, W1r, b1, hbuf, n);
    }
    // 6) scatter h into agg2 (d=32)
    {
        long threads = (long)e * 8;
        scatter_k<<<(int)((threads + 255) / 256), 256, 0, stream>>>(hbuf, src, dst, agg2, e, 32);
    }
    // 7) layer 2 (WMMA) + dropout + row L2 normalize
    {
        int tiles = (n + 15) / 16;
        sage2_k<<<(tiles + 7) / 8, 256, 0, stream>>>(hbuf, agg2, rinv, W2l, W2r, b2, out, n);
    }
}